// BCGrounder_28630251995231
// MI455X (gfx1250) — compile-verified
//
#include <hip/hip_runtime.h>
#include <stdint.h>

#define P_CONST 40
#define E_CONST 1024
#define TABLE_ELEMS (P_CONST * E_CONST * E_CONST)   // 41,943,040 floats = 167.8 MB (fits 192MB L2)

// ---------------- Kernel 1: zero the table (reference starts from jnp.zeros) ----------------
__global__ void BCG_zero_kernel(float4* __restrict__ t, int n4) {
    int i = blockIdx.x * blockDim.x + threadIdx.x;
    const int stride = gridDim.x * blockDim.x;
    const float4 z = make_float4(0.f, 0.f, 0.f, 0.f);
    for (; i < n4; i += stride) t[i] = z;
}

// ---------------- Kernel 2: scatter-max facts into table ----------------
// Scores are uniform [0,1) and table is 0-initialized -> all values >= 0,
// so float ordering == unsigned-bit ordering: use GLOBAL_ATOMIC_MAX_U32.
__global__ void BCG_scatter_kernel(const int* __restrict__ fp,
                                   const int* __restrict__ fa0,
                                   const int* __restrict__ fa1,
                                   const float* __restrict__ fs,
                                   unsigned* __restrict__ table, int F) {
    int i = blockIdx.x * blockDim.x + threadIdx.x;
    if (i >= F) return;
    const int p  = fp[i];
    const int a0 = fa0[i];
    const int a1 = fa1[i];
    const unsigned bits = __float_as_uint(fs[i]);
    const size_t idx = ((size_t)p << 20) + ((size_t)a0 << 10) + (size_t)a1;
    atomicMax(table + idx, bits);
}

// ---------------- Kernel 3: one block per query ----------------
// direct = table[qp, qa0, qa1]
// best   = max over rules with head==qp, over k of  w[r] * table[b1,qa0,k] * table[b2,k,qa1]
// Row table[b1,qa0,:] (4KB contiguous) is staged into LDS with the CDNA5 async
// global->LDS path (global_load_async_to_lds_b128 + s_wait_asynccnt).
__global__ __launch_bounds__(256) void BCG_query_kernel(
        const float* __restrict__ table,
        const int*   __restrict__ rh,  const int* __restrict__ rb1,
        const int*   __restrict__ rb2, const float* __restrict__ rw,
        const int*   __restrict__ qp,  const int* __restrict__ qa0,
        const int*   __restrict__ qa1, float* __restrict__ out, int R) {
    __shared__ __align__(16) float srow[E_CONST];   // 4KB row buffer
    __shared__ float wred[8];                        // per-wave partial maxima

    const int q   = blockIdx.x;
    const int tid = threadIdx.x;
    const int pq = qp[q];
    const int a0 = qa0[q];
    const int a1 = qa1[q];

    float direct = 0.f;
    if (tid == 0)
        direct = table[((size_t)pq << 20) + ((size_t)a0 << 10) + (size_t)a1];

    // LDS byte address of this lane's 16B chunk (flat LDS addr truncates to offset)
    const unsigned lds_off = (unsigned)(uintptr_t)(&srow[0]) + (unsigned)(tid * 16);
    const unsigned goff    = (unsigned)(tid * 16);

    float tbest = 0.f;   // running max of w[r]*s1[k]*s2[k] over this thread's (r,k)
    for (int r = 0; r < R; ++r) {
        if (rh[r] != pq) continue;               // uniform across the block

        const float* rowbase = table + (((size_t)rb1[r] << 20) + ((size_t)a0 << 10));

        __syncthreads();                          // previous row fully consumed
        // CDNA5 async copy: 256 lanes x 16B = 4KB row -> LDS (tracked by ASYNCcnt)
        asm volatile("global_load_async_to_lds_b128 %0, %1, %2"
                     :
                     : "v"(lds_off), "v"(goff), "s"(rowbase)
                     : "memory");
        asm volatile("s_wait_asynccnt 0" ::: "memory");
        __syncthreads();                          // all waves' chunks visible

        const float4 s1 = *reinterpret_cast<const float4*>(&srow[tid * 4]);
        const float  w  = rw[r];
        const float* colbase = table + (((size_t)rb2[r] << 20) + (size_t)a1);
        const int k = tid * 4;
        const float p0 = s1.x * colbase[(size_t)(k + 0) << 10];
        const float p1 = s1.y * colbase[(size_t)(k + 1) << 10];
        const float p2 = s1.z * colbase[(size_t)(k + 2) << 10];
        const float p3 = s1.w * colbase[(size_t)(k + 3) << 10];
        const float m  = fmaxf(fmaxf(p0, p1), fmaxf(p2, p3));
        tbest = fmaxf(tbest, w * m);              // w>=0: scaling commutes with max
    }

    // wave32 max-reduction, then combine the 8 waves via LDS
    #pragma unroll
    for (int off = 16; off > 0; off >>= 1)
        tbest = fmaxf(tbest, __shfl_xor(tbest, off, 32));
    const int wid  = tid >> 5;
    const int lane = tid & 31;
    if (lane == 0) wred[wid] = tbest;
    __syncthreads();
    if (tid == 0) {
        float b = wred[0];
        #pragma unroll
        for (int i = 1; i < 8; ++i) b = fmaxf(b, wred[i]);
        out[q] = fmaxf(direct, b);
    }
}

extern "C" void kernel_launch(void* const* d_in, const int* in_sizes, int n_in,
                              void* d_out, int out_size, void* d_ws, size_t ws_size,
                              hipStream_t stream) {
    const int*   fact_pred    = (const int*)  d_in[0];
    const int*   fact_a0      = (const int*)  d_in[1];
    const int*   fact_a1      = (const int*)  d_in[2];
    const float* fact_scores  = (const float*)d_in[3];
    const int*   rules_head   = (const int*)  d_in[4];
    const int*   rules_b1     = (const int*)  d_in[5];
    const int*   rules_b2     = (const int*)  d_in[6];
    const float* rule_weights = (const float*)d_in[7];
    const int*   query_pred   = (const int*)  d_in[8];
    const int*   query_a0     = (const int*)  d_in[9];
    const int*   query_a1     = (const int*)  d_in[10];
    float* out   = (float*)d_out;
    float* table = (float*)d_ws;   // 167.8 MB scratch table

    const int F = in_sizes[0];
    const int R = in_sizes[4];
    const int Q = in_sizes[8];

    // 1) zero table (must happen every call: ws is not re-poisoned/cleared)
    const int n4 = TABLE_ELEMS / 4;
    BCG_zero_kernel<<<4096, 256, 0, stream>>>((float4*)table, n4);

    // 2) scatter-max facts
    BCG_scatter_kernel<<<(F + 255) / 256, 256, 0, stream>>>(
        fact_pred, fact_a0, fact_a1, fact_scores, (unsigned*)table, F);

    // 3) evaluate queries, one 256-thread (8-wave) block per query
    BCG_query_kernel<<<Q, 256, 0, stream>>>(
        table, rules_head, rules_b1, rules_b2, rule_weights,
        query_pred, query_a0, query_a1, out, R);
}